// RNN_64252710748591
// MI455X (gfx1250) — compile-verified
//
#include <hip/hip_runtime.h>

// LSTM: B=8192, T=512, I=12, H=20; gates per PyTorch order (i,f,g,o); out = Linear(h_T) -> [B,10]
// Strategy: 1 wave32 per 16-batch tile, loop T; fp32 WMMA 16x16x4 for both x@W_ih^T and h@W_hh^T,
// weights resident in VGPRs, gates/h staged through wave-private LDS, c-state in registers.
// Pointwise cell uses CDNA5 hardware tanh (sigmoid via 0.5*tanh(x/2)+0.5) -- no IEEE divides.

#define TLEN 512
#define IDIM 12
#define HDIM 20
#define G4   80
#define ODIM 10

typedef __attribute__((ext_vector_type(2))) float v2f;
typedef __attribute__((ext_vector_type(8))) float v8f;

__device__ __forceinline__ float fast_rcp(float x) {
#if __has_builtin(__builtin_amdgcn_rcpf)
  return __builtin_amdgcn_rcpf(x);        // v_rcp_f32
#else
  return 1.0f / x;
#endif
}

__device__ __forceinline__ float fast_tanh(float x) {
#if __has_builtin(__builtin_amdgcn_tanhf)
  return __builtin_amdgcn_tanhf(x);       // v_tanh_f32 (CDNA5 trans op)
#else
  // tanh(x) = 2*sigmoid(2x) - 1, sigmoid via exp + rcp (avoids IEEE divide)
  const float e = __expf(-2.0f * x);
  return __builtin_fmaf(-2.0f, e * fast_rcp(1.0f + e), 1.0f);
#endif
}

__device__ __forceinline__ float fast_sigmoid(float x) {
  // exact identity: sigmoid(x) = 0.5*tanh(x/2) + 0.5
  return __builtin_fmaf(0.5f, fast_tanh(0.5f * x), 0.5f);
}

__global__ __launch_bounds__(32) void lstm_wmma_f32(
    const float* __restrict__ x,      // [8192, 512, 12]
    const float* __restrict__ w_ih,   // [80, 12]
    const float* __restrict__ w_hh,   // [80, 20]
    const float* __restrict__ b_ih,   // [80]
    const float* __restrict__ b_hh,   // [80]
    const float* __restrict__ w_out,  // [10, 20]
    const float* __restrict__ b_out,  // [10]
    float* __restrict__ out)          // [8192, 10]
{
  __shared__ __align__(16) float g_lds[16 * G4];   // gates tile [16][80]
  __shared__ __align__(16) float h_lds[16 * HDIM]; // hidden tile [16][20]

  const int lane = threadIdx.x;   // 0..31
  const int half = lane >> 4;     // 0/1 (K split in A/B fragments)
  const int col  = lane & 15;
  const int b0   = blockIdx.x * 16;

  // ---- resident weight fragments (B matrices: K x 16 tiles, row striped across lanes) ----
  v2f   bh[5][5];   // w_hh^T : [kchunk][ntile], B[k][n] = w_hh[n][k]
  v2f   bi[3][5];   // w_ih^T
  float biasf[5];   // (b_ih + b_hh) per N tile, indexed by lane%16
  #pragma unroll
  for (int nt = 0; nt < 5; ++nt) {
    const int n = nt * 16 + col;
    #pragma unroll
    for (int kc = 0; kc < 5; ++kc) {
      const int kb = kc * 4 + half * 2;          // 0..18, even
      bh[kc][nt].x = w_hh[n * HDIM + kb];
      bh[kc][nt].y = w_hh[n * HDIM + kb + 1];
    }
    #pragma unroll
    for (int kc = 0; kc < 3; ++kc) {
      const int kb = kc * 4 + half * 2;          // 0..10, even
      bi[kc][nt].x = w_ih[n * IDIM + kb];
      bi[kc][nt].y = w_ih[n * IDIM + kb + 1];
    }
    biasf[nt] = b_ih[n] + b_hh[n];
  }

  // ---- per-lane pointwise element mapping: e = lane + 32r -> (m = e/20, j = e%20) ----
  int mr[10], jr[10];
  #pragma unroll
  for (int r = 0; r < 10; ++r) {
    const int e = lane + 32 * r;
    mr[r] = e / HDIM;
    jr[r] = e - mr[r] * HDIM;
  }

  // ---- zero initial h tile and c state ----
  float cst[10];
  #pragma unroll
  for (int r = 0; r < 10; ++r) {
    cst[r] = 0.0f;
    h_lds[lane + 32 * r] = 0.0f;
  }
  __syncthreads();

  const float* xrow = x + (size_t)(b0 + col) * TLEN * IDIM;

  for (int t = 0; t < TLEN; ++t) {
    // keep the x stream ahead of the serial WMMA chain (speculative, RT)
    if (t + 8 < TLEN) __builtin_prefetch(xrow + (t + 8) * IDIM, 0, 0);

    // A fragments: x tile 16x12 (3 K-chunks), h tile 16x20 (5 K-chunks).
    // 16x4 f32 A layout: lane(col) = row M, VGPR pair holds K = 4*kc + 2*half + {0,1} -> contiguous 8B.
    v2f ax[3];
    #pragma unroll
    for (int kc = 0; kc < 3; ++kc) {
      const int kb = kc * 4 + half * 2;
      ax[kc] = *(const v2f*)(xrow + t * IDIM + kb);   // 8B-aligned global_load_b64
    }
    v2f ah[5];
    #pragma unroll
    for (int kc = 0; kc < 5; ++kc) {
      const int kb = kc * 4 + half * 2;
      ah[kc] = *(const v2f*)(&h_lds[col * HDIM + kb]); // 8B-aligned ds_load_b64
    }

    // gates[16][80] = bias + x W_ih^T + h W_hh^T, five independent 16-wide N tiles
    #pragma unroll
    for (int nt = 0; nt < 5; ++nt) {
      v8f acc;
      #pragma unroll
      for (int v = 0; v < 8; ++v) acc[v] = biasf[nt];
      #pragma unroll
      for (int kc = 0; kc < 3; ++kc)
        acc = __builtin_amdgcn_wmma_f32_16x16x4_f32(
            false, ax[kc], false, bi[kc][nt], (short)0, acc, false, false);
      #pragma unroll
      for (int kc = 0; kc < 5; ++kc)
        acc = __builtin_amdgcn_wmma_f32_16x16x4_f32(
            false, ah[kc], false, bh[kc][nt], (short)0, acc, false, false);
      // C/D layout: VGPR v, lanes<16 -> M=v, lanes>=16 -> M=v+8; N = nt*16 + col
      #pragma unroll
      for (int v = 0; v < 8; ++v)
        g_lds[(v + 8 * half) * G4 + nt * 16 + col] = acc[v];
    }
    __syncthreads();

    // pointwise LSTM cell: 320 (m,j) cells, 10 per lane; c stays in registers
    #pragma unroll
    for (int r = 0; r < 10; ++r) {
      const int m = mr[r], j = jr[r];
      const float ig = g_lds[m * G4 + j];
      const float fg = g_lds[m * G4 + 20 + j];
      const float gg = g_lds[m * G4 + 40 + j];
      const float og = g_lds[m * G4 + 60 + j];
      const float i_ = fast_sigmoid(ig);
      const float f_ = fast_sigmoid(fg);
      const float g_ = fast_tanh(gg);
      const float o_ = fast_sigmoid(og);
      const float c  = __builtin_fmaf(f_, cst[r], i_ * g_);
      cst[r] = c;
      h_lds[m * HDIM + j] = o_ * fast_tanh(c);
    }
    __syncthreads();
  }

  // ---- final linear: out[m][o] = sum_j h[m][j] * w_out[o][j] + b_out[o]; 160 elems, 5/lane ----
  #pragma unroll
  for (int r = 0; r < 5; ++r) {
    const int e  = lane + 32 * r;     // < 160
    const int m  = e / ODIM;
    const int oo = e - m * ODIM;
    float s = b_out[oo];
    #pragma unroll
    for (int j = 0; j < HDIM; ++j)
      s = __builtin_fmaf(h_lds[m * HDIM + j], w_out[oo * HDIM + j], s);
    out[(size_t)(b0 + m) * ODIM + oo] = s;
  }
}

extern "C" void kernel_launch(void* const* d_in, const int* in_sizes, int n_in,
                              void* d_out, int out_size, void* d_ws, size_t ws_size,
                              hipStream_t stream) {
  const float* x     = (const float*)d_in[0];
  const float* w_ih  = (const float*)d_in[1];
  const float* w_hh  = (const float*)d_in[2];
  const float* b_ih  = (const float*)d_in[3];
  const float* b_hh  = (const float*)d_in[4];
  const float* w_out = (const float*)d_in[5];
  const float* b_out = (const float*)d_in[6];
  float* out = (float*)d_out;

  const int B = 8192;
  dim3 grid(B / 16);   // 512 one-wave workgroups
  dim3 block(32);
  lstm_wmma_f32<<<grid, block, 0, stream>>>(x, w_ih, w_hh, b_ih, b_hh, w_out, b_out, out);
}